// flow_module_10316511445570
// MI455X (gfx1250) — compile-verified
//
#include <hip/hip_runtime.h>
#include <math.h>

#define HH 512
#define WW 512
#define NCH 48
#define NQ (NCH / 4)          /* 12 float4 chunks per pixel */
#define HWPIX (HH * WW)
#define TS 16
#define TP (TS + 2)           /* 18 */
#define TPIX (TP * TP)        /* 324 */

// ---------------------------------------------------------------------------
// Kernel 1: per-pixel 3x3 stencil coefficients from Dt (periodic rolls),
// with conv zero-padding folded in as zeroed taps and 1/dg folded in.
// Output: 9 SoA planes of HWPIX floats.
// ---------------------------------------------------------------------------
__global__ __launch_bounds__(256) void compute_A_kernel(
    const float* __restrict__ Dt, const float* __restrict__ dg,
    float* __restrict__ A)
{
    int pix = blockIdx.x * blockDim.x + threadIdx.x;
    if (pix >= HWPIX) return;
    int h = pix >> 9, w = pix & (WW - 1);

    float a00,c00,b00, c01,b01, a10,b10, b11;
    float c0_1,b0_1, a_10,b_10, b_1_1, b_11, b1_1;
    {
        const float* p;
        p = Dt + ((((h    ) & (HH-1)) << 9) | ((w    ) & (WW-1))) * 3; a00=p[0]; c00=p[1]; b00=p[2];
        p = Dt + ((((h    ) & (HH-1)) << 9) | ((w - 1) & (WW-1))) * 3; c01=p[1]; b01=p[2];
        p = Dt + ((((h - 1) & (HH-1)) << 9) | ((w    ) & (WW-1))) * 3; a10=p[0]; b10=p[2];
        p = Dt + ((((h - 1) & (HH-1)) << 9) | ((w - 1) & (WW-1))) * 3; b11=p[2];
        p = Dt + ((((h    ) & (HH-1)) << 9) | ((w + 1) & (WW-1))) * 3; c0_1=p[1]; b0_1=p[2];
        p = Dt + ((((h + 1) & (HH-1)) << 9) | ((w    ) & (WW-1))) * 3; a_10=p[0]; b_10=p[2];
        p = Dt + ((((h + 1) & (HH-1)) << 9) | ((w + 1) & (WW-1))) * 3; b_1_1=p[2];
        p = Dt + ((((h + 1) & (HH-1)) << 9) | ((w - 1) & (WW-1))) * 3; b_11=p[2];
        p = Dt + ((((h - 1) & (HH-1)) << 9) | ((w + 1) & (WW-1))) * 3; b1_1=p[2];
    }

    float A0 = (fabsf(b_11) - b_11 + fabsf(b00) - b00) * 0.25f;
    float A1 = (c01 + c00 - fabsf(b01) - fabsf(b00)) * 0.5f;
    float A2 = (fabsf(b11) + b11 + fabsf(b00) + b00) * 0.25f;
    float A3 = (a_10 + a00 - fabsf(b_10) - fabsf(b00)) * 0.5f;
    float A4 = -(a_10 + 2.0f * a00 + a10) * 0.5f
             - (fabsf(b_11) - b_11 + fabsf(b11) + b11) * 0.25f
             - (fabsf(b_1_1) + b_1_1 + fabsf(b1_1) - b1_1) * 0.25f
             + (fabsf(b_10) + fabsf(b10) + fabsf(b0_1) + fabsf(b01) + 2.0f * fabsf(b00)) * 0.5f
             - (c0_1 + 2.0f * c00 + c01) * 0.5f;
    float A5 = (a10 + a00 - fabsf(b10) - fabsf(b00)) * 0.5f;
    float A6 = (fabsf(b_1_1) + b_1_1 + fabsf(b00) + b00) * 0.25f;
    float A7 = (c0_1 + c00 - fabsf(b0_1) - fabsf(b00)) * 0.5f;
    float A8 = (fabsf(b1_1) - b1_1 + fabsf(b00) - b00) * 0.25f;

    // Fold conv zero padding: tap k = 3*i + j touches (h+i-1, w+j-1)
    bool top = (h == 0), bot = (h == HH - 1), lft = (w == 0), rgt = (w == WW - 1);
    if (top || lft) A0 = 0.0f;
    if (top)        A1 = 0.0f;
    if (top || rgt) A2 = 0.0f;
    if (lft)        A3 = 0.0f;
    if (rgt)        A5 = 0.0f;
    if (bot || lft) A6 = 0.0f;
    if (bot)        A7 = 0.0f;
    if (bot || rgt) A8 = 0.0f;

    float inv = 1.0f / dg[pix];   // fold LB(...)/dg into the weights
    A[0 * HWPIX + pix] = A0 * inv;
    A[1 * HWPIX + pix] = A1 * inv;
    A[2 * HWPIX + pix] = A2 * inv;
    A[3 * HWPIX + pix] = A3 * inv;
    A[4 * HWPIX + pix] = A4 * inv;
    A[5 * HWPIX + pix] = A5 * inv;
    A[6 * HWPIX + pix] = A6 * inv;
    A[7 * HWPIX + pix] = A7 * inv;
    A[8 * HWPIX + pix] = A8 * inv;
}

// ---------------------------------------------------------------------------
// Kernel 2 (fused Euler step): 16x16 pixel tile per block.
//  1) async-stage the 18x18x48 halo tile into LDS (periodic wrap values;
//     conv zero-padding already folded into A),
//  2) cooperatively compute logsumexp for all 324 tile pixels in LDS,
//  3) per-thread pixel: 9-tap per-pixel stencil + log_softmax forward
//     differences + quadratic form + Pi_0 + clip + Euler update.
// Single pass over x per step.
// ---------------------------------------------------------------------------
__global__ __launch_bounds__(256) void step_kernel(
    const float* __restrict__ x, const float* __restrict__ A,
    const float* __restrict__ hinv, float* __restrict__ out)
{
    __shared__ float4 tile[TPIX][NQ];   // 18*18*48*4 = 62208 B
    __shared__ float  lse_s[TPIX];      // + 1296 B   = 63504 B total

    const int tid = threadIdx.y * TS + threadIdx.x;
    const int h0 = blockIdx.y * TS, w0 = blockIdx.x * TS;

    // ---- (1) async stage of the halo tile into LDS (16B per lane per op) ----
    for (int idx = tid; idx < TPIX * NQ; idx += 256) {
        int p  = idx / NQ;
        int q  = idx - p * NQ;
        int py = p / TP, px = p - py * TP;
        int gh = (h0 + py - 1) & (HH - 1);
        int gw = (w0 + px - 1) & (WW - 1);
        const float* src = x + ((size_t)((gh << 9) | gw)) * NCH + q * 4;
        unsigned lds_off = (unsigned)(unsigned long long)(size_t)&tile[p][q];
        unsigned long long gaddr = (unsigned long long)(size_t)src;
        asm volatile("global_load_async_to_lds_b128 %0, %1, off"
                     :: "v"(lds_off), "v"(gaddr) : "memory");
    }
    asm volatile("s_wait_asynccnt 0" ::: "memory");
    __syncthreads();

    // ---- (2) logsumexp for every tile pixel (wrapped values => wrapped lse) ----
    for (int p = tid; p < TPIX; p += 256) {
        float m = tile[p][0].x;
#pragma unroll
        for (int q = 0; q < NQ; q++) {
            float4 r = tile[p][q];
            m = fmaxf(m, fmaxf(fmaxf(r.x, r.y), fmaxf(r.z, r.w)));
        }
        float s = 0.0f;
#pragma unroll
        for (int q = 0; q < NQ; q++) {
            float4 r = tile[p][q];
            s += __expf(r.x - m) + __expf(r.y - m) +
                 __expf(r.z - m) + __expf(r.w - m);
        }
        lse_s[p] = m + __logf(s);
    }
    __syncthreads();

    // ---- (3) per-thread pixel ----
    const int py = threadIdx.y, px = threadIdx.x;
    const int h = h0 + py, w = w0 + px;
    const int gidx = (h << 9) | w;

    float sA[9];
#pragma unroll
    for (int k = 0; k < 9; k++) sA[k] = A[k * HWPIX + gidx];

    int nb[9];
#pragma unroll
    for (int i = 0; i < 3; i++)
#pragma unroll
        for (int j = 0; j < 3; j++) nb[3 * i + j] = (py + i) * TP + (px + j);
    const int c0 = nb[4];

    const float lc  = lse_s[c0];
    const float dld = lse_s[nb[7]] - lc;   // lse(h+1,w) - lse(h,w)
    const float dlr = lse_s[nb[5]] - lc;   // lse(h,w+1) - lse(h,w)

    const float* hv = hinv + (size_t)gidx * 3;
    const float q0 = 0.5f * hv[0];   // 0.5 * hinv_a  (du0^2)
    const float q1 = 0.5f * hv[1];   // 0.5 * hinv_c  (du1^2)
    const float q2 = hv[2];          // hinv_b        (2*b*0.5 cross)

    float4 val[NQ];
    float ssum = 0.0f;

#pragma unroll
    for (int q = 0; q < NQ; q++) {
        float4 L  = {0.f, 0.f, 0.f, 0.f};
        float4 xc = {0.f, 0.f, 0.f, 0.f};
        float4 xd = {0.f, 0.f, 0.f, 0.f};
        float4 xr = {0.f, 0.f, 0.f, 0.f};
#pragma unroll
        for (int k = 0; k < 9; k++) {
            float4 xn = tile[nb[k]][q];
            if (k == 4) xc = xn;       // (h, w)
            if (k == 5) xr = xn;       // (h, w+1)
            if (k == 7) xd = xn;       // (h+1, w)
            float a = sA[k];
            L.x = fmaf(a, xn.x, L.x);
            L.y = fmaf(a, xn.y, L.y);
            L.z = fmaf(a, xn.z, L.z);
            L.w = fmaf(a, xn.w, L.w);
        }
        // du0 = logp(h+1,w)-logp(h,w) = (xd-xc) - dld ; du1 likewise with w+1
        float d0x = (xd.x - xc.x) - dld, d1x = (xr.x - xc.x) - dlr;
        float d0y = (xd.y - xc.y) - dld, d1y = (xr.y - xc.y) - dlr;
        float d0z = (xd.z - xc.z) - dld, d1z = (xr.z - xc.z) - dlr;
        float d0w = (xd.w - xc.w) - dld, d1w = (xr.w - xc.w) - dlr;
        float4 o;
        o.x = L.x + q0 * d0x * d0x + q2 * d0x * d1x + q1 * d1x * d1x + 0.1f * xc.x;
        o.y = L.y + q0 * d0y * d0y + q2 * d0y * d1y + q1 * d1y * d1y + 0.1f * xc.y;
        o.z = L.z + q0 * d0z * d0z + q2 * d0z * d1z + q1 * d1z * d1z + 0.1f * xc.z;
        o.w = L.w + q0 * d0w * d0w + q2 * d0w * d1w + q1 * d1w * d1w + 0.1f * xc.w;
        val[q] = o;
        ssum += o.x + o.y + o.z + o.w;
    }

    const float mean = ssum * (1.0f / 48.0f);
    float4* op = (float4*)(out + (size_t)gidx * NCH);
#pragma unroll
    for (int q = 0; q < NQ; q++) {
        float4 xc = tile[c0][q];
        float tx = fminf(fmaxf(val[q].x - mean, -1e8f), 1e8f);
        float ty = fminf(fmaxf(val[q].y - mean, -1e8f), 1e8f);
        float tz = fminf(fmaxf(val[q].z - mean, -1e8f), 1e8f);
        float tw = fminf(fmaxf(val[q].w - mean, -1e8f), 1e8f);
        float4 o;
        o.x = fmaf(0.2f, tx, xc.x);
        o.y = fmaf(0.2f, ty, xc.y);
        o.z = fmaf(0.2f, tz, xc.z);
        o.w = fmaf(0.2f, tw, xc.w);
        op[q] = o;
    }
}

// ---------------------------------------------------------------------------
extern "C" void kernel_launch(void* const* d_in, const int* in_sizes, int n_in,
                              void* d_out, int out_size, void* d_ws, size_t ws_size,
                              hipStream_t stream)
{
    const float* v    = (const float*)d_in[0];
    const float* Dt   = (const float*)d_in[1];
    const float* dg   = (const float*)d_in[2];
    const float* hinv = (const float*)d_in[3];
    float* out = (float*)d_out;

    float* A  = (float*)d_ws;             // 9 * HWPIX floats
    float* xt = A + 9 * HWPIX;            // HWPIX * NCH floats (ping buffer)

    dim3 b1(256), g1((HWPIX + 255) / 256);
    compute_A_kernel<<<g1, b1, 0, stream>>>(Dt, dg, A);

    dim3 bs(TS, TS), gs(WW / TS, HH / TS);
    const float* cur = v;
    for (int s = 0; s < 5; s++) {
        float* dst = (s & 1) ? xt : out;  // v->out->xt->out->xt->out
        step_kernel<<<gs, bs, 0, stream>>>(cur, A, hinv, dst);
        cur = dst;
    }
}